// SatLossEvaluator_31353261260819
// MI455X (gfx1250) — compile-verified
//
#include <hip/hip_runtime.h>

// ---------------------------------------------------------------------------
// SAT loss evaluator for MI455X (gfx1250, wave32).
//
// Edge pass streams graph_map / edge_feature tiles into LDS with the Tensor
// Data Mover (tensor_load_to_lds, TENSORcnt), double-buffered so the DMA of
// tile k+1 overlaps gather+exp+atomic work on tile k. Frees the VMEM queues
// for the random vp[] gather (4MB, L2-hot) and the 24M float atomics into the
// 32MB accumulator region (L2-resident on the 192MB L2).
//
// ws layout:
//   [0,   8)                : double accumulator for sum of log(clause_value)
//   [8,  16)                : pad
//   [16, 16 + 8*n_clauses)  : float2 acc[c] = { sum w*v, sum w }
// ---------------------------------------------------------------------------

#define BLOCK 256
#define TILE  2048                     // edges per LDS tile (8KB per array)
#define WS_ACC_OFFSET_FLOATS 4

typedef __attribute__((ext_vector_type(4))) unsigned int u32x4;
typedef __attribute__((ext_vector_type(8))) int          i32x8;
typedef __attribute__((ext_vector_type(4))) int          i32x4;

// Issue one TDM 1-D tile load: tile_elems 4-byte elements from gptr -> LDS
// offset lds_off. n_valid = elements remaining in the tensor from gptr
// (hardware OOB check zero-fills reads past it, handling the tail tile).
// D# layout per CDNA5 ISA.
__device__ inline void tdm_load_tile(unsigned lds_off, const void* gptr,
                                     unsigned n_valid, unsigned tile_elems) {
    unsigned long long ga = (unsigned long long)(uintptr_t)gptr;
    u32x4 g0 = {
        1u,                                            // count=1, normal mode
        lds_off,                                       // lds_addr (bytes)
        (unsigned)(ga & 0xFFFFFFFFu),                  // global_addr[31:0]
        (unsigned)((ga >> 32) & 0x01FFFFFFu) | (2u << 30) // addr[56:32] | type=2
    };
    i32x8 g1 = {
        (int)(2u << 16),                               // wg_mask=0, data_size=4B
        (int)((n_valid & 0xFFFFu) << 16),              // tensor_dim0[15:0]
        (int)(((n_valid >> 16) & 0xFFFFu) | (1u << 16)), // dim0[31:16] | tensor_dim1=1
        (int)(tile_elems << 16),                       // tile_dim0
        1,                                             // tile_dim1=1, tile_dim2=0
        0, 0, 0                                        // strides unused (1 row)
    };
    i32x4 gz = {0, 0, 0, 0};                           // groups 2/3: <=2D tensor
#if __clang_major__ >= 23
    i32x8 gz8 = {0, 0, 0, 0, 0, 0, 0, 0};
    __builtin_amdgcn_tensor_load_to_lds(g0, g1, gz, gz, gz8, 0);
#else
    __builtin_amdgcn_tensor_load_to_lds(g0, g1, gz, gz, 0);
#endif
}

// Zero the accumulator region (n_clauses is a device scalar).
__global__ __launch_bounds__(BLOCK)
void sat_zero_ws(float* __restrict__ ws, const int* __restrict__ n_clauses_p) {
    const int n = *n_clauses_p;
    const long long total  = (long long)WS_ACC_OFFSET_FLOATS + 2LL * n;
    const long long stride = (long long)gridDim.x * blockDim.x;
    for (long long i = (long long)blockIdx.x * blockDim.x + threadIdx.x;
         i < total; i += stride)
        ws[i] = 0.0f;
}

// Edge pass: TDM double-buffered tiles of (lits, clauses, ef) in LDS;
// per-edge gather vp[lit], w = exp(coeff*v), scatter-add via L2 atomics.
__global__ __launch_bounds__(BLOCK)
void sat_edge_pass(const float* __restrict__ vp,
                   const int*   __restrict__ lits,
                   const int*   __restrict__ clauses,
                   const float* __restrict__ ef,
                   const float* __restrict__ global_step,
                   float2*      __restrict__ acc,
                   int n_edges) {
    __shared__ int   s_lit[2][TILE];
    __shared__ int   s_cls[2][TILE];
    __shared__ float s_ef [2][TILE];

    const float coeff = fminf(__builtin_sqrtf(global_step[0]), 10.0f);
    const int n_tiles = (n_edges + TILE - 1) / TILE;

    int t = (int)blockIdx.x;
    // Preload first tile (wave 0 only: TDM ignores EXEC, so gate per-wave).
    if (t < n_tiles && threadIdx.x < 32) {
        long long s = (long long)t * TILE;
        unsigned rem = (unsigned)(n_edges - s);
        tdm_load_tile((unsigned)(uintptr_t)&s_lit[0][0], lits    + s, rem, TILE);
        tdm_load_tile((unsigned)(uintptr_t)&s_cls[0][0], clauses + s, rem, TILE);
        tdm_load_tile((unsigned)(uintptr_t)&s_ef [0][0], ef      + s, rem, TILE);
    }

    int buf = 0;
    for (; t < n_tiles; t += (int)gridDim.x) {
        // Wave 0's TENSORcnt covers this tile's DMA; barrier publishes LDS.
        __builtin_amdgcn_s_wait_tensorcnt(0);
        __syncthreads();

        // Kick off next tile into the other buffer (overlaps compute below).
        const int tn = t + (int)gridDim.x;
        if (tn < n_tiles && threadIdx.x < 32) {
            long long s = (long long)tn * TILE;
            unsigned rem = (unsigned)(n_edges - s);
            const int nb = buf ^ 1;
            tdm_load_tile((unsigned)(uintptr_t)&s_lit[nb][0], lits    + s, rem, TILE);
            tdm_load_tile((unsigned)(uintptr_t)&s_cls[nb][0], clauses + s, rem, TILE);
            tdm_load_tile((unsigned)(uintptr_t)&s_ef [nb][0], ef      + s, rem, TILE);
        }

        // Process current tile.
        const long long base = (long long)t * TILE;
        #pragma unroll 2
        for (int i = threadIdx.x; i < TILE; i += BLOCK) {
            const long long e = base + i;
            if (e < n_edges) {
                const int   lit = s_lit[buf][i];
                const int   cl  = s_cls[buf][i];
                const float f   = s_ef [buf][i];
                const float v   = f * vp[lit] + (1.0f - f) * 0.5f;
                const float w   = __expf(coeff * v);     // v_exp_f32
                atomicAdd(&acc[cl].x, w * v);            // nominator
                atomicAdd(&acc[cl].y, w);                // denominator
            }
        }
        __syncthreads();   // all reads of buf done before it is re-filled
        buf ^= 1;
    }
}

// Clause pass: cv = den / max(nom, eps); sharpen; block-reduce log;
// one f64 atomic per block. wave32 shuffle widths.
__global__ __launch_bounds__(BLOCK)
void sat_clause_pass(const float2* __restrict__ acc,
                     double*       __restrict__ result_acc,
                     const int*    __restrict__ n_clauses_p,
                     const float*  __restrict__ eps_p) {
    const int   n      = *n_clauses_p;
    const float eps    = eps_p[0];
    const int   stride = gridDim.x * blockDim.x;

    float local = 0.0f;
    for (int c = blockIdx.x * blockDim.x + threadIdx.x; c < n; c += stride) {
        const float2 a = acc[c];
        float cv = a.y / fmaxf(a.x, eps);
        const float d  = cv - 1.0f;
        const float d2 = d * d;
        cv = 1.0f + d2 * d2 * d;                         // 1 + (cv-1)^5
        local += __logf(fmaxf(cv, eps));                 // v_log_f32
    }

    #pragma unroll
    for (int off = 16; off > 0; off >>= 1)
        local += __shfl_down(local, off, 32);

    __shared__ float wsum[BLOCK / 32];
    const int lane = threadIdx.x & 31;
    const int wid  = threadIdx.x >> 5;
    if (lane == 0) wsum[wid] = local;
    __syncthreads();

    if (wid == 0) {
        local = (lane < (BLOCK / 32)) ? wsum[lane] : 0.0f;
        #pragma unroll
        for (int off = 16; off > 0; off >>= 1)
            local += __shfl_down(local, off, 32);
        if (lane == 0)
            atomicAdd(result_acc, (double)local);        // global_atomic_add_f64
    }
}

__global__ void sat_finalize(const double* __restrict__ result_acc,
                             const int*    __restrict__ n_clauses_p,
                             float*        __restrict__ out) {
    if (blockIdx.x == 0 && threadIdx.x == 0)
        out[0] = (float)(*result_acc / (double)(*n_clauses_p));
}

extern "C" void kernel_launch(void* const* d_in, const int* in_sizes, int n_in,
                              void* d_out, int out_size, void* d_ws, size_t ws_size,
                              hipStream_t stream) {
    const float* vp          = (const float*)d_in[0];
    const int*   gmap        = (const int*)  d_in[1];   // [2, n_edges]
    const float* ef          = (const float*)d_in[2];
    const int*   n_clauses_p = (const int*)  d_in[3];   // device scalar
    const float* global_step = (const float*)d_in[4];
    const float* eps_p       = (const float*)d_in[5];

    const int n_edges  = in_sizes[2];
    const int* lits    = gmap;
    const int* clauses = gmap + n_edges;

    float*  ws_f       = (float*)d_ws;
    double* result_acc = (double*)d_ws;
    float2* acc        = (float2*)(ws_f + WS_ACC_OFFSET_FLOATS);

    // 1) zero workspace accumulators (every call: graph-replay safe).
    sat_zero_ws<<<4096, BLOCK, 0, stream>>>(ws_f, n_clauses_p);

    // 2) edge scatter pass: TDM-staged tiles, grid-strided over tiles.
    const int n_tiles = (n_edges + TILE - 1) / TILE;
    int edge_blocks = n_tiles < 1536 ? (n_tiles < 1 ? 1 : n_tiles) : 1536;
    sat_edge_pass<<<edge_blocks, BLOCK, 0, stream>>>(
        vp, lits, clauses, ef, global_step, acc, n_edges);

    // 3) clause reduce pass.
    sat_clause_pass<<<2048, BLOCK, 0, stream>>>(acc, result_acc, n_clauses_p, eps_p);

    // 4) finalize mean into d_out.
    sat_finalize<<<1, 32, 0, stream>>>(result_acc, n_clauses_p, (float*)d_out);
}